// GCN_17540646437275
// MI455X (gfx1250) — compile-verified
//
#include <hip/hip_runtime.h>
#include <cstdint>

// fp32 WMMA operand types (wave32): A/B 16x4 -> 2 VGPRs, C/D 16x16 -> 8 VGPRs
typedef __attribute__((ext_vector_type(2))) float v2f;
typedef __attribute__((ext_vector_type(8))) float v8f;

// GCC-style 16-byte int vector: matches the async-to-LDS builtin's parameter type
// ('__attribute__((__vector_size__(4 * sizeof(int)))) int *' per hipcc diagnostic).
typedef int v4i_vs __attribute__((vector_size(16)));

// ---------- gfx1250 async global->LDS copy (ASYNCcnt path), with safe fallback ----------
#if defined(__has_builtin)
#if __has_builtin(__builtin_amdgcn_global_load_async_to_lds_b128)
#define HAVE_ASYNC_LDS 1
#endif
#endif
#ifndef HAVE_ASYNC_LDS
#define HAVE_ASYNC_LDS 0
#endif

__device__ __forceinline__ void tile_to_lds(const float* __restrict__ g, float* l,
                                            int nfloats, int tid, int nthreads) {
#if HAVE_ASYNC_LDS
  typedef __attribute__((address_space(1))) v4i_vs gv4i;  // global int4
  typedef __attribute__((address_space(3))) v4i_vs lv4i;  // LDS int4
  for (int c = tid * 4; c < nfloats; c += nthreads * 4) {
    __builtin_amdgcn_global_load_async_to_lds_b128(
        (gv4i*)(g + c), (lv4i*)(l + c), /*offset=*/0, /*cpol=*/0);
  }
#if __has_builtin(__builtin_amdgcn_s_wait_asynccnt)
  __builtin_amdgcn_s_wait_asynccnt(0);
#else
  asm volatile("s_wait_asynccnt 0x0" ::: "memory");
#endif
#else
  for (int c = tid * 4; c < nfloats; c += nthreads * 4) {
    float4 v = *(const float4*)(g + c);
    *(float4*)(l + c) = v;
  }
#endif
}

// ---------------- degree / normalization ----------------
__global__ void k_init_deg(float* deg, int n) {
  int i = blockIdx.x * blockDim.x + threadIdx.x;
  if (i < n) deg[i] = 1.0f;  // self loop counts once
}

__global__ void k_count_deg(const int* __restrict__ dst, float* deg, int e) {
  int i = blockIdx.x * blockDim.x + threadIdx.x;
  if (i < e) atomicAdd(&deg[dst[i]], 1.0f);
}

__global__ void k_dinv(float* deg, int n) {
  int i = blockIdx.x * blockDim.x + threadIdx.x;
  if (i < n) {
    float d = deg[i];
    deg[i] = d > 0.0f ? rsqrtf(d) : 0.0f;
  }
}

// ---------------- dense GEMM via V_WMMA_F32_16X16X4_F32, LDS-staged ----------------
// C[M,NOUT] = (RELU ? relu(A) : A)[M,K] @ B[K,NOUT]
//  - whole B (K*NOUT floats) staged in LDS once per block (async-to-LDS when available)
//  - grid-stride loop over 16-row A tiles; each tile (16*K floats) staged in LDS
//  - wave w owns output columns [w*16, w*16+16); K fully unrolled -> 32 WMMAs
// Operand layouts (wave32):
//   A 16x4 (MxK):  m = lane&15;  VGPR0/1 hold K = 2*(lane>>4) + {0,1}
//   B 4x16 (KxN):  n = lane&15;  VGPR0/1 hold K = 2*(lane>>4) + {0,1}
//   C/D 16x16 f32: VGPR r: lanes 0-15 -> row r, lanes 16-31 -> row r+8
template <int K, int NOUT, bool RELU>
__global__ void k_gemm_wmma(const float* __restrict__ A, const float* __restrict__ B,
                            float* __restrict__ C, int ntm) {
  extern __shared__ float smem[];
  float* sB = smem;            // K*NOUT floats
  float* sA = smem + K * NOUT; // 16*K floats

  const int tid  = threadIdx.x;
  const int bd   = blockDim.x;     // 32 * (NOUT/16)
  const int wave = tid >> 5;       // == tn
  const int lane = tid & 31;
  const int half = lane >> 4;      // 0 or 1
  const int l15  = lane & 15;
  const int n    = wave * 16 + l15;

  // stage weights once per block
  tile_to_lds(B, sB, K * NOUT, tid, bd);
  __syncthreads();

  for (int tm = blockIdx.x; tm < ntm; tm += gridDim.x) {
    __syncthreads();  // previous iteration finished reading sA
    tile_to_lds(A + (size_t)tm * 16 * K, sA, 16 * K, tid, bd);
    __syncthreads();  // tile visible to all waves

    const float* arow = sA + l15 * K;
    v8f acc = {};
#pragma unroll
    for (int k0 = 0; k0 < K; k0 += 4) {
      int ka = k0 + 2 * half;
      v2f av = *(const v2f*)(arow + ka);  // ds_load_b64, 8B aligned
      if (RELU) {
        av.x = av.x > 0.0f ? av.x : 0.0f;
        av.y = av.y > 0.0f ? av.y : 0.0f;
      }
      v2f bv;
      bv.x = sB[ka * NOUT + n];
      bv.y = sB[(ka + 1) * NOUT + n];
      acc = __builtin_amdgcn_wmma_f32_16x16x4_f32(
          /*neg_a=*/false, av, /*neg_b=*/false, bv,
          /*c_mod=*/(short)0, acc, /*reuse_a=*/false, /*reuse_b=*/false);
    }

    float* crow = C + (size_t)(tm * 16 + half * 8) * NOUT + n;
#pragma unroll
    for (int r = 0; r < 8; ++r)
      crow[(size_t)r * NOUT] = acc[r];
  }
}

// ---------------- aggregation ----------------
// out[i,:] = bias + tmp[i,:] * dinv[i]^2   (self-loop term, full overwrite)
__global__ void k_agg_init(const float* __restrict__ tmp, const float* __restrict__ dinv,
                           const float* __restrict__ bias, float* __restrict__ out,
                           int n, int f) {
  int fv = f >> 2;
  int t = blockIdx.x * blockDim.x + threadIdx.x;
  if (t >= n * fv) return;
  int node = t / fv;
  int j = t - node * fv;
  float w = dinv[node];
  w *= w;
  float4 tv = ((const float4*)(tmp + (size_t)node * f))[j];
  float4 bv = ((const float4*)bias)[j];
  float4 o;
  o.x = bv.x + tv.x * w;
  o.y = bv.y + tv.y * w;
  o.z = bv.z + tv.z * w;
  o.w = bv.w + tv.w * w;
  ((float4*)(out + (size_t)node * f))[j] = o;
}

// out[dst,:] += tmp[src,:] * dinv[src]*dinv[dst]; one wave == one edge (f=128),
// float4 gather per lane + 4 global_atomic_add_f32 scatter (L2-resident tables).
__global__ void k_agg_edges(const float* __restrict__ tmp, const float* __restrict__ dinv,
                            const int* __restrict__ src, const int* __restrict__ dst,
                            float* __restrict__ out, int e, int f) {
  int fv = f >> 2;
  long long t = (long long)blockIdx.x * blockDim.x + threadIdx.x;
  if (t >= (long long)e * fv) return;
  int edge = (int)(t / fv);
  int j = (int)(t - (long long)edge * fv);
  int s = src[edge], d = dst[edge];  // wave-uniform for f=128
  float w = dinv[s] * dinv[d];
  float4 tv = ((const float4*)(tmp + (size_t)s * f))[j];
  float* op = out + (size_t)d * f + (size_t)j * 4;
  atomicAdd(op + 0, tv.x * w);
  atomicAdd(op + 1, tv.y * w);
  atomicAdd(op + 2, tv.z * w);
  atomicAdd(op + 3, tv.w * w);
}

// ---------------- final [N,64] @ [64,1] + bo ----------------
__global__ void k_out(const float* __restrict__ z, const float* __restrict__ Wo,
                      const float* __restrict__ bo, float* __restrict__ out, int n) {
  int wave = threadIdx.x >> 5;
  int lane = threadIdx.x & 31;
  int node = blockIdx.x * (blockDim.x >> 5) + wave;
  if (node >= n) return;
  const float* zr = z + (size_t)node * 64;
  float s = zr[lane] * Wo[lane] + zr[lane + 32] * Wo[lane + 32];
#pragma unroll
  for (int off = 16; off > 0; off >>= 1) s += __shfl_xor(s, off, 32);
  if (lane == 0) out[node] = s + bo[0];
}

extern "C" void kernel_launch(void* const* d_in, const int* in_sizes, int n_in,
                              void* d_out, int out_size, void* d_ws, size_t ws_size,
                              hipStream_t stream) {
  const float* x  = (const float*)d_in[0];
  const int*   ei = (const int*)d_in[1];
  const float* W1 = (const float*)d_in[2];
  const float* b1 = (const float*)d_in[3];
  const float* Wh = (const float*)d_in[4];
  const float* bh = (const float*)d_in[5];
  const float* W2 = (const float*)d_in[6];
  const float* b2 = (const float*)d_in[7];
  const float* Wo = (const float*)d_in[8];
  const float* bo = (const float*)d_in[9];
  float* out = (float*)d_out;

  const int N = 100000, E = 1600000, H = 128, D = 64;
  const int* src = ei;      // edge_index[0]
  const int* dst = ei + E;  // edge_index[1]

  // workspace: deg/dinv [N] | tmp [N*H] | bufA [N*H]   (~103 MB)
  float* deg  = (float*)d_ws;
  float* tmp  = deg + N;
  float* bufA = tmp + (size_t)N * H;

  // --- degree -> dinv (in place) ---
  k_init_deg <<<(N + 255) / 256, 256, 0, stream>>>(deg, N);
  k_count_deg<<<(E + 255) / 256, 256, 0, stream>>>(dst, deg, E);
  k_dinv     <<<(N + 255) / 256, 256, 0, stream>>>(deg, N);

  const int ntm = N / 16;        // 6250 row tiles, exact
  const int gemm_blocks = 625;   // 10 tiles per block; W staged once per block
  const size_t sh128 = (size_t)(128 * 128 + 16 * 128) * sizeof(float);  // 72 KB
  const size_t sh64  = (size_t)(128 * 64  + 16 * 128) * sizeof(float);  // 40 KB

  // --- layer 1: tmp = x@W1 ; bufA = agg(tmp)+b1 (relu deferred into next GEMM) ---
  {
    k_gemm_wmma<128, 128, false><<<gemm_blocks, 256, sh128, stream>>>(x, W1, tmp, ntm);
    int ti = N * (H / 4);
    k_agg_init<<<(ti + 255) / 256, 256, 0, stream>>>(tmp, deg, b1, bufA, N, H);
    long long te = (long long)E * (H / 4);
    k_agg_edges<<<(unsigned)((te + 255) / 256), 256, 0, stream>>>(tmp, deg, src, dst, bufA, E, H);
  }
  // --- layer 2: tmp = relu(bufA)@Wh ; bufA = agg(tmp)+bh ---
  {
    k_gemm_wmma<128, 128, true><<<gemm_blocks, 256, sh128, stream>>>(bufA, Wh, tmp, ntm);
    int ti = N * (H / 4);
    k_agg_init<<<(ti + 255) / 256, 256, 0, stream>>>(tmp, deg, bh, bufA, N, H);
    long long te = (long long)E * (H / 4);
    k_agg_edges<<<(unsigned)((te + 255) / 256), 256, 0, stream>>>(tmp, deg, src, dst, bufA, E, H);
  }
  // --- layer 3: tmp = relu(bufA)@W2 ; bufA = agg(tmp)+b2 (no relu) ---
  {
    k_gemm_wmma<128, 64, true><<<gemm_blocks, 128, sh64, stream>>>(bufA, W2, tmp, ntm);
    int ti = N * (D / 4);
    k_agg_init<<<(ti + 255) / 256, 256, 0, stream>>>(tmp, deg, b2, bufA, N, D);
    long long te = (long long)E * (D / 4);
    k_agg_edges<<<(unsigned)((te + 255) / 256), 256, 0, stream>>>(tmp, deg, src, dst, bufA, E, D);
  }
  // --- out = bufA @ Wo + bo ---
  k_out<<<(N + 7) / 8, 256, 0, stream>>>(bufA, Wo, bo, out, N);
}